// HFlow_40492951667019
// MI455X (gfx1250) — compile-verified
//
#include <hip/hip_runtime.h>
#include <hip/hip_bf16.h>

typedef float v4f __attribute__((ext_vector_type(4)));
typedef int   v4i __attribute__((ext_vector_type(4)));

typedef __attribute__((address_space(1))) v4i gas_v4i;   // global
typedef __attribute__((address_space(3))) v4i las_v4i;   // LDS

#define TPB   256          // 8 wave32s per block
#define LROW  2048         // row length (elements)
#define WAVES (TPB / 32)

#if defined(__gfx1250__) && __has_builtin(__builtin_amdgcn_global_load_async_to_lds_b128)
#define HAS_ASYNC_LDS 1
#else
#define HAS_ASYNC_LDS 0
#endif

__global__ __launch_bounds__(TPB) void
HFlow_40492951667019_kernel(const float* __restrict__ v,
                            const float* __restrict__ z,
                            float* __restrict__ out)
{
    __shared__ float zbuf[LROW];        // 8 KB: this row's z staged via async DMA
    __shared__ float red_vz[WAVES];
    __shared__ float red_vv[WAVES];

    const int    t    = threadIdx.x;
    const size_t base = (size_t)blockIdx.x * LROW;

    const v4f* __restrict__ vrow = (const v4f*)(v + base);
    v4f*       __restrict__ orow = (v4f*)(out + base);

    // ---- Kick off async global->LDS stream of z (2 x b128 per lane) ----
#if HAS_ASYNC_LDS
    {
        gas_v4i* gz = (gas_v4i*)(z + base);     // cast away const for DMA src
        las_v4i* lz = (las_v4i*)zbuf;
        __builtin_amdgcn_global_load_async_to_lds_b128(gz + t,       lz + t,       0, 0);
        __builtin_amdgcn_global_load_async_to_lds_b128(gz + 256 + t, lz + 256 + t, 0, 0);
    }
#else
    {
        const v4f* zrow = (const v4f*)(z + base);
        ((v4f*)zbuf)[t]       = __builtin_nontemporal_load(zrow + t);
        ((v4f*)zbuf)[t + 256] = __builtin_nontemporal_load(zrow + t + 256);
    }
#endif

    // ---- Overlap: stream v into registers (NT) and start v.v ----
    v4f a0 = __builtin_nontemporal_load(vrow + t);
    v4f a1 = __builtin_nontemporal_load(vrow + t + 256);

    float vv = 0.0f;
    vv = fmaf(a0.x, a0.x, vv); vv = fmaf(a0.y, a0.y, vv);
    vv = fmaf(a0.z, a0.z, vv); vv = fmaf(a0.w, a0.w, vv);
    vv = fmaf(a1.x, a1.x, vv); vv = fmaf(a1.y, a1.y, vv);
    vv = fmaf(a1.z, a1.z, vv); vv = fmaf(a1.w, a1.w, vv);

    // ---- Wait for this wave's async z transfers; each wave reads only the
    //      LDS region its own lanes wrote, so no block barrier is needed. ----
#if HAS_ASYNC_LDS
#if __has_builtin(__builtin_amdgcn_s_wait_asynccnt)
    __builtin_amdgcn_s_wait_asynccnt(0);
#else
    asm volatile("s_wait_asynccnt 0" ::: "memory");
#endif
#endif

    v4f z0 = ((const v4f*)zbuf)[t];
    v4f z1 = ((const v4f*)zbuf)[t + 256];

    float vz = 0.0f;
    vz = fmaf(a0.x, z0.x, vz); vz = fmaf(a0.y, z0.y, vz);
    vz = fmaf(a0.z, z0.z, vz); vz = fmaf(a0.w, z0.w, vz);
    vz = fmaf(a1.x, z1.x, vz); vz = fmaf(a1.y, z1.y, vz);
    vz = fmaf(a1.z, z1.z, vz); vz = fmaf(a1.w, z1.w, vz);

    // ---- Wave32 butterfly reduction ----
    #pragma unroll
    for (int off = 16; off >= 1; off >>= 1) {
        vz += __shfl_xor(vz, off, 32);
        vv += __shfl_xor(vv, off, 32);
    }

    const int wave = t >> 5;
    if ((t & 31) == 0) {
        red_vz[wave] = vz;
        red_vv[wave] = vv;
    }
    __syncthreads();

    float tvz = 0.0f, tvv = 0.0f;
    #pragma unroll
    for (int w = 0; w < WAVES; ++w) {
        tvz += red_vz[w];
        tvv += red_vv[w];
    }

    const float s = 2.0f * tvz / tvv;

    v4f o0, o1;
    o0.x = fmaf(-s, a0.x, z0.x); o0.y = fmaf(-s, a0.y, z0.y);
    o0.z = fmaf(-s, a0.z, z0.z); o0.w = fmaf(-s, a0.w, z0.w);
    o1.x = fmaf(-s, a1.x, z1.x); o1.y = fmaf(-s, a1.y, z1.y);
    o1.z = fmaf(-s, a1.z, z1.z); o1.w = fmaf(-s, a1.w, z1.w);

    __builtin_nontemporal_store(o0, orow + t);
    __builtin_nontemporal_store(o1, orow + t + 256);
}

extern "C" void kernel_launch(void* const* d_in, const int* in_sizes, int n_in,
                              void* d_out, int out_size, void* d_ws, size_t ws_size,
                              hipStream_t stream) {
    (void)n_in; (void)d_ws; (void)ws_size;
    const float* v  = (const float*)d_in[0];
    const float* z  = (const float*)d_in[1];
    float* out      = (float*)d_out;

    const int rows = out_size / LROW;   // B = 16384 for the reference shapes
    (void)in_sizes;
    HFlow_40492951667019_kernel<<<rows, TPB, 0, stream>>>(v, z, out);
}